// KSpaceResampling_7447473291349
// MI455X (gfx1250) — compile-verified
//
#include <hip/hip_runtime.h>
#include <math.h>

// ---------------------------------------------------------------------------
// NUDFT resampling (type-2 -> density comp -> type-1 adjoint -> abs)
// FP32 WMMA (v_wmma_f32_16x16x4_f32) on gfx1250, wave32.
// ---------------------------------------------------------------------------

typedef __attribute__((ext_vector_type(2))) float v2f;
typedef __attribute__((ext_vector_type(8))) float v8f;

#define WMMA_F32(A, B, C) \
  __builtin_amdgcn_wmma_f32_16x16x4_f32(false, (A), false, (B), (short)0, (C), false, false)

namespace {
constexpr int   H      = 192;
constexpr int   W      = 192;
constexpr int   HHALF  = 96;            // H // 2
constexpr int   NVIEWS = 128;
constexpr int   NSAMP  = 384;
constexpr int   M      = NVIEWS * NSAMP;  // 49152
constexpr float INV_HW = 1.0f / (192.0f * 192.0f);
}

// ---------------------------------------------------------------------------
// Kernel 1: type-2 NUDFT + row-dot with Ay + density weights  -> k[cb][m]
// Block: 128 threads (4 waves). Block owns 16 m's; wave w owns image cb = w.
// ---------------------------------------------------------------------------
__global__ __launch_bounds__(128) void nudft_forward_kernel(
    const float* __restrict__ x,        // [B=2, H, W, C=2]
    const float* __restrict__ points,   // [M, 2]
    const float* __restrict__ weights,  // [M]
    float* __restrict__ kbuf)           // [4, M, 2]  (workspace)
{
  __shared__ float axr[16][192], axi[16][192];   // exp(-i kx n): cos, -sin
  __shared__ float ayr[16][192], ayi[16][192];   // exp(-i ky n): cos, -sin

  const int m0 = blockIdx.x * 16;
  const int t  = threadIdx.x;

  // --- build phase tables via recurrence: job t -> (m_local, h-segment of 24)
  {
    const int   m_l = t & 15;
    const int   seg = t >> 4;
    const int   h_s = seg * 24;
    const float kx  = points[2 * (m0 + m_l) + 0];
    const float ky  = points[2 * (m0 + m_l) + 1];
    float cc, ss, cd, sd;

    __sincosf(kx * (float)(h_s - HHALF), &ss, &cc);
    __sincosf(kx, &sd, &cd);
    #pragma unroll
    for (int i = 0; i < 24; ++i) {
      axr[m_l][h_s + i] = cc;
      axi[m_l][h_s + i] = -ss;
      const float c2 = cc * cd - ss * sd;
      ss = ss * cd + cc * sd;
      cc = c2;
    }
    __sincosf(ky * (float)(h_s - HHALF), &ss, &cc);
    __sincosf(ky, &sd, &cd);
    #pragma unroll
    for (int i = 0; i < 24; ++i) {
      ayr[m_l][h_s + i] = cc;
      ayi[m_l][h_s + i] = -ss;
      const float c2 = cc * cd - ss * sd;
      ss = ss * cd + cc * sd;
      cc = c2;
    }
  }
  __syncthreads();

  const int cb   = t >> 5;          // wave id == image index
  const int c    = cb & 1;
  const int b    = cb >> 1;
  const int lane = t & 31;
  const int half = lane >> 4;
  const int l16  = lane & 15;

  float kr[8], ki[8];
  #pragma unroll
  for (int j = 0; j < 8; ++j) { kr[j] = 0.0f; ki[j] = 0.0f; }

  for (int nt = 0; nt < 12; ++nt) {
    const int w0   = nt * 16;
    const int wcol = w0 + l16;
    v8f accR = {};
    v8f accI = {};
    for (int k0 = 0; k0 < H; k0 += 4) {
      const int kk = k0 + 2 * half;       // A/B frag K pair for this lane half
      v2f Ar, Ai, Bf;
      Ar.x = axr[l16][kk];     Ar.y = axr[l16][kk + 1];
      Ai.x = axi[l16][kk];     Ai.y = axi[l16][kk + 1];
      // image is purely real (xc = complex(x, 0)): B frag from global (L2-hot)
      Bf.x = x[((b * H + kk    ) * W + wcol) * 2 + c];
      Bf.y = x[((b * H + kk + 1) * W + wcol) * 2 + c];
      accR = WMMA_F32(Ar, Bf, accR);      // tmp_re
      accI = WMMA_F32(Ai, Bf, accI);      // tmp_im
    }
    // fold Ay element-wise on the 16x16 tile, accumulate per-lane partial k[m]
    #pragma unroll
    for (int j = 0; j < 8; ++j) {
      const int   m_l = j + 8 * half;     // C layout: reg j -> M=j / M=j+8
      const float tr  = accR[j];
      const float ti  = accI[j];
      const float yr  = ayr[m_l][wcol];
      const float yi  = ayi[m_l][wcol];
      kr[j] += tr * yr - ti * yi;
      ki[j] += tr * yi + ti * yr;
    }
  }

  // reduce over the 16 lanes inside each half (xor stays within the half)
  #pragma unroll
  for (int j = 0; j < 8; ++j) {
    #pragma unroll
    for (int off = 1; off < 16; off <<= 1) {
      kr[j] += __shfl_xor(kr[j], off, 32);
      ki[j] += __shfl_xor(ki[j], off, 32);
    }
  }
  if (l16 == 0) {
    #pragma unroll
    for (int j = 0; j < 8; ++j) {
      const int   m   = m0 + j + 8 * half;
      const float wgt = weights[m];
      kbuf[(cb * M + m) * 2 + 0] = kr[j] * wgt;
      kbuf[(cb * M + m) * 2 + 1] = ki[j] * wgt;
    }
  }
}

// ---------------------------------------------------------------------------
// Kernel 2: type-1 adjoint NUDFT + abs.  Grid (12,12,4): one 16x16 tile of
// one image per block.  4 waves split the staged 128-m chunk (deterministic),
// partials combined through LDS in fixed order.
// ---------------------------------------------------------------------------
__global__ __launch_bounds__(128) void nudft_adjoint_kernel(
    const float* __restrict__ points,   // [M, 2]
    const float* __restrict__ kbuf,     // [4, M, 2]
    float* __restrict__ out)            // [B, H, W, C]
{
  __shared__ float ur[128][16], ui[128][16];   // conj(Ax): exp(+i kx n)
  __shared__ float vr[128][16], vi[128][16];   // conj(Ay): exp(+i ky n)
  __shared__ float kvr[128], kvi[128];
  __shared__ float redR[4][256], redI[4][256];

  const int w0   = blockIdx.x * 16;
  const int h0   = blockIdx.y * 16;
  const int img  = blockIdx.z;
  const int c    = img & 1;
  const int b    = img >> 1;
  const int t    = threadIdx.x;
  const int wave = t >> 5;
  const int lane = t & 31;
  const int half = lane >> 4;
  const int l16  = lane & 15;

  v8f accR = {};
  v8f accI = {};

  for (int r = 0; r < M / 128; ++r) {
    const int mb = r * 128;
    // --- stage phases + k for 128 samples (recurrence: 4 sincos per thread)
    {
      const int   m  = mb + t;
      const float kx = points[2 * m + 0];
      const float ky = points[2 * m + 1];
      float cc, ss, cd, sd;

      __sincosf(kx * (float)(h0 - HHALF), &ss, &cc);
      __sincosf(kx, &sd, &cd);
      #pragma unroll
      for (int i = 0; i < 16; ++i) {
        ur[t][i] = cc;
        ui[t][i] = ss;
        const float c2 = cc * cd - ss * sd;
        ss = ss * cd + cc * sd;
        cc = c2;
      }
      __sincosf(ky * (float)(w0 - HHALF), &ss, &cc);
      __sincosf(ky, &sd, &cd);
      #pragma unroll
      for (int i = 0; i < 16; ++i) {
        vr[t][i] = cc;
        vi[t][i] = ss;
        const float c2 = cc * cd - ss * sd;
        ss = ss * cd + cc * sd;
        cc = c2;
      }
      kvr[t] = kbuf[(img * M + m) * 2 + 0];
      kvi[t] = kbuf[(img * M + m) * 2 + 1];
    }
    __syncthreads();

    // --- wave handles local-m range [wave*32, wave*32+32): 8 K-steps of 4
    const int base = wave * 32 + 2 * half;
    #pragma unroll
    for (int s = 0; s < 8; ++s) {
      const int mA = base + s * 4;    // frag .x local-m index; .y = mA+1
      v2f Ar, Ai, nAi, Br, Bi;
      {
        const float k0r = kvr[mA],     k0i = kvi[mA];
        const float u0r = ur[mA][l16], u0i = ui[mA][l16];
        Ar.x = k0r * u0r - k0i * u0i;           // p_re = (k * u).re
        Ai.x = k0r * u0i + k0i * u0r;           // p_im = (k * u).im
        const float k1r = kvr[mA + 1],     k1i = kvi[mA + 1];
        const float u1r = ur[mA + 1][l16], u1i = ui[mA + 1][l16];
        Ar.y = k1r * u1r - k1i * u1i;
        Ai.y = k1r * u1i + k1i * u1r;
        nAi.x = -Ai.x;
        nAi.y = -Ai.y;
        Br.x = vr[mA][l16];     Br.y = vr[mA + 1][l16];
        Bi.x = vi[mA][l16];     Bi.y = vi[mA + 1][l16];
      }
      // img_re += pr*vr - pi*vi ; img_im += pr*vi + pi*vr
      accR = WMMA_F32(Ar,  Br, accR);
      accR = WMMA_F32(nAi, Bi, accR);
      accI = WMMA_F32(Ar,  Bi, accI);
      accI = WMMA_F32(Ai,  Br, accI);
    }
    __syncthreads();
  }

  // --- deterministic fixed-order combine of the 4 waves' partial tiles
  #pragma unroll
  for (int j = 0; j < 8; ++j) {
    redR[wave][j * 32 + lane] = accR[j];
    redI[wave][j * 32 + lane] = accI[j];
  }
  __syncthreads();

  #pragma unroll
  for (int e2 = 0; e2 < 2; ++e2) {
    const int   e  = t + e2 * 128;
    const float sr = redR[0][e] + redR[1][e] + redR[2][e] + redR[3][e];
    const float si = redI[0][e] + redI[1][e] + redI[2][e] + redI[3][e];
    const int   j  = e >> 5;
    const int   L  = e & 31;
    const int   hh = h0 + j + 8 * (L >> 4);   // C layout inverse map
    const int   ww = w0 + (L & 15);
    out[((b * H + hh) * W + ww) * 2 + c] = sqrtf(sr * sr + si * si) * INV_HW;
  }
}

// ---------------------------------------------------------------------------
extern "C" void kernel_launch(void* const* d_in, const int* in_sizes, int n_in,
                              void* d_out, int out_size, void* d_ws, size_t ws_size,
                              hipStream_t stream) {
  (void)in_sizes; (void)n_in; (void)out_size; (void)ws_size;
  const float* x       = (const float*)d_in[0];   // [2,192,192,2] f32
  const float* points  = (const float*)d_in[1];   // [49152, 2]    f32
  const float* weights = (const float*)d_in[2];   // [49152]       f32
  float*       out     = (float*)d_out;           // [2,192,192,2] f32
  float*       kbuf    = (float*)d_ws;            // 4*M*2 floats = 1.5 MB

  nudft_forward_kernel<<<dim3(M / 16), dim3(128), 0, stream>>>(x, points, weights, kbuf);
  nudft_adjoint_kernel<<<dim3(12, 12, 4), dim3(128), 0, stream>>>(points, kbuf, out);
}